// LatentSpaceTransformer_62517543961241
// MI455X (gfx1250) — compile-verified
//
#include <hip/hip_runtime.h>
#include <hip/hip_bf16.h>

// Problem geometry (fixed by the reference).
#define H_    256
#define W_    256
#define C_    32
#define NVIEW 16
#define NNOV  4
#define HW_   (H_ * W_)

#define GLOBAL_AS __attribute__((address_space(1)))
#define LDS_AS    __attribute__((address_space(3)))

// ---- CDNA5 async global->LDS probe (gfx1250) --------------------------------
#if defined(__has_builtin)
#  if __has_builtin(__builtin_amdgcn_global_load_async_to_lds_b32)
#    define HAVE_ASYNC_LDS 1
#  endif
#  if __has_builtin(__builtin_amdgcn_s_wait_asynccnt)
#    define HAVE_WAIT_ASYNC_BUILTIN 1
#  endif
#endif
#ifndef HAVE_ASYNC_LDS
#  define HAVE_ASYNC_LDS 0
#endif
#ifndef HAVE_WAIT_ASYNC_BUILTIN
#  define HAVE_WAIT_ASYNC_BUILTIN 0
#endif

__device__ __forceinline__ void wait_asynccnt0() {
#if HAVE_WAIT_ASYNC_BUILTIN
    __builtin_amdgcn_s_wait_asynccnt(0);
#else
    asm volatile("s_wait_asynccnt 0" ::: "memory");
#endif
}

// One block per (n, v, y) output row. 256 threads = 8 wave32 waves, thread = x.
__global__ __launch_bounds__(W_) void mpi_warp_kernel(
    const float* __restrict__ feats,   // [16, 32, 256, 256]
    const int*   __restrict__ depth,   // [16, 1, 256, 256]
    const float* __restrict__ novel,   // [4, 2]  (x, y)
    const float* __restrict__ vpos,    // [16, 2] (x, y)
    float*       __restrict__ out)     // [4, 16, 32, 256, 256]
{
    __shared__ int sdepth[4 * W_];     // 4 candidate depth rows, 4 KB

    const int gid = blockIdx.x;
    const int y   = gid & (H_ - 1);
    const int v   = (gid >> 8) & (NVIEW - 1);
    const int n   = gid >> 12;
    const int x   = threadIdx.x;

    // disp[l] = round(((vpos - novel) * diop_l) * (F_CAM*FEAT_DIM)) — exact
    // left-to-right multiply order of the reference; rintf == round-half-even.
    const float dxf = vpos[2 * v + 0] - novel[2 * n + 0];
    const float dyf = vpos[2 * v + 1] - novel[2 * n + 1];
    const float diop[4] = {0.5f, 1.0f, 2.0f, 3.0f};

    int  dxl[4], syl[4];
    bool rowok[4];
#pragma unroll
    for (int l = 0; l < 4; ++l) {
        dxl[l]   = (int)rintf((dxf * diop[l]) * 128.0f);
        const int dy = (int)rintf((dyf * diop[l]) * 128.0f);
        syl[l]   = y - dy;                    // dy constant over the row
        rowok[l] = (syl[l] >= 0) && (syl[l] < H_);
    }

    // Stage the four candidate depth rows into LDS (coalesced, one b32/lane
    // per layer). gfx1250 path: async DMA global->LDS tracked by ASYNCcnt.
#pragma unroll
    for (int l = 0; l < 4; ++l) {
        if (rowok[l]) {
            const int* g = depth + ((size_t)v * H_ + (size_t)syl[l]) * W_ + x;
#if HAVE_ASYNC_LDS
            __builtin_amdgcn_global_load_async_to_lds_b32(
                (GLOBAL_AS int*)g,
                (LDS_AS int*)&sdepth[l * W_ + x],
                0, 0);
#else
            sdepth[l * W_ + x] = *g;
#endif
        }
    }
#if HAVE_ASYNC_LDS
    wait_asynccnt0();
#endif
    __syncthreads();

    // Winner = highest layer whose (in-bounds) source pixel has depth == l.
    int win = -1, wsx = 0, wsy = 0;
#pragma unroll
    for (int l = 0; l < 4; ++l) {
        const int sx = x - dxl[l];
        if (rowok[l] && sx >= 0 && sx < W_ && sdepth[l * W_ + sx] == l) {
            win = l; wsx = sx; wsy = syl[l];
        }
    }

    // Copy all 32 channels (stride HW between channel planes). Output is a
    // 512 MB stream never re-read -> non-temporal stores keep feats hot in L2.
    const size_t dst = (((size_t)n * NVIEW + v) * C_) * HW_ + (size_t)y * W_ + x;
    if (win >= 0) {
        const float* src = feats + ((size_t)v * C_) * HW_
                                 + (size_t)wsy * W_ + (size_t)wsx;
        __builtin_prefetch(src, 0, 0);        // global_prefetch_b8
#pragma unroll
        for (int c = 0; c < C_; ++c) {
            const float val = src[(size_t)c * HW_];
            __builtin_nontemporal_store(val, &out[dst + (size_t)c * HW_]);
        }
    } else {
#pragma unroll
        for (int c = 0; c < C_; ++c) {
            __builtin_nontemporal_store(0.0f, &out[dst + (size_t)c * HW_]);
        }
    }
}

extern "C" void kernel_launch(void* const* d_in, const int* in_sizes, int n_in,
                              void* d_out, int out_size, void* d_ws, size_t ws_size,
                              hipStream_t stream) {
    (void)in_sizes; (void)n_in; (void)out_size; (void)d_ws; (void)ws_size;
    const float* feats = (const float*)d_in[0];
    const int*   depth = (const int*)d_in[1];
    const float* novel = (const float*)d_in[2];
    const float* vpos  = (const float*)d_in[3];
    float*       out   = (float*)d_out;

    const dim3 grid(NNOV * NVIEW * H_);   // 16384 blocks: one per (n, v, y)
    const dim3 block(W_);                 // 256 threads = 8 wave32 waves
    mpi_warp_kernel<<<grid, block, 0, stream>>>(feats, depth, novel, vpos, out);
}